// RegionLoss_Ava_13838384627856
// MI455X (gfx1250) — compile-verified
//
#include <hip/hip_runtime.h>
#include <hip/hip_bf16.h>

// ---------------- problem constants (match reference setup_inputs) ----------
#define NB   32
#define NAA  3
#define NC   80
#define NH   56
#define NW   56
#define NT   50                       // targets per image
#define PLANE (NH * NW)               // 3136
#define CELLS (NAA * PLANE)           // 9408 cells per image
#define BLOCK 224                     // 7 wave32s; 42 * 224 == 9408 exactly
#define BLOCKS_PER_B (CELLS / BLOCK)  // 42
#define EPSF 1e-7f

typedef float v2f __attribute__((ext_vector_type(2)));
typedef float v8f __attribute__((ext_vector_type(8)));

__device__ __forceinline__ float sigmoidf_(float v) { return 1.0f / (1.0f + expf(-v)); }

// Exact-fp32 full-wave32 sum using V_WMMA_F32_16X16X4_F32.
// A: lane L reg0 = partial p_L, reg1 = 0  (A[m,0]=p_m, A[m,1]=0, A[m,2]=p_{m+16}, A[m,3]=0)
// B: all ones  =>  D[m,n] = p_m + p_{m+16} for every n.
// C/D layout: lanes 0-15 hold rows M=0..7 in d[0..7]; lanes 16-31 hold M=8..15.
// sum(d[0..7]) gives half-wave sums; one xor-16 shuffle completes the wave sum.
__device__ __forceinline__ float wave_sum_wmma(float v) {
    v2f a; a[0] = v;    a[1] = 0.0f;
    v2f b; b[0] = 1.0f; b[1] = 1.0f;
    v8f c = {};
    v8f d = __builtin_amdgcn_wmma_f32_16x16x4_f32(
        /*neg_a=*/false, a, /*neg_b=*/false, b,
        /*c_mod=*/(short)0, c, /*reuse_a=*/false, /*reuse_b=*/false);
    float s = ((d[0] + d[1]) + (d[2] + d[3])) + ((d[4] + d[5]) + (d[6] + d[7]));
    s += __shfl_xor(s, 16, 32);
    return s;   // every lane holds the full 32-lane sum
}

// ---------------- kernel A: per-target precompute (1600 items) --------------
// tgt8[g] = {gx, gy, gw, gh, tx, ty, tw, th}; tflat[g] = cell index in image or -1
__global__ void rl_targets(const float* __restrict__ tboxes,
                           float* __restrict__ tgt8, int* __restrict__ tflat) {
    int g = blockIdx.x * blockDim.x + threadIdx.x;
    if (g >= NB * NT) return;
    const float aw[3] = {1.28967f, 2.12714f, 3.27212f};
    const float ah[3] = {4.15014f, 5.09344f, 5.87423f};
    float cx = tboxes[g * 4 + 0], cy = tboxes[g * 4 + 1];
    float tw_ = tboxes[g * 4 + 2], th_ = tboxes[g * 4 + 3];
    float gx = cx * (float)NW, gy = cy * (float)NH;
    float gw = tw_ * (float)NW, gh = th_ * (float)NH;
    // anchor argmax (first-wins on ties, matching jnp.argmax)
    int   bn = 0; float best = -1.0f;
    #pragma unroll
    for (int a = 0; a < 3; ++a) {
        float inter = fminf(aw[a], gw) * fminf(ah[a], gh);
        float r = inter / (aw[a] * ah[a] + gw * gh - inter);
        if (r > best) { best = r; bn = a; }
    }
    int gi = min(max((int)gx, 0), NW - 1);
    int gj = min(max((int)gy, 0), NH - 1);
    tgt8[g * 8 + 0] = gx;
    tgt8[g * 8 + 1] = gy;
    tgt8[g * 8 + 2] = gw;
    tgt8[g * 8 + 3] = gh;
    tgt8[g * 8 + 4] = gx - (float)gi;
    tgt8[g * 8 + 5] = gy - (float)gj;
    tgt8[g * 8 + 6] = logf(gw / aw[bn]);
    tgt8[g * 8 + 7] = logf(gh / ah[bn]);
    tflat[g] = (tw_ > 0.0f) ? ((bn * NH + gj) * NW + gi) : -1;
}

// ---------------- init: zero the 3 global accumulators ----------------------
__global__ void rl_init(float* __restrict__ acc) {
    acc[0] = 0.0f; acc[1] = 0.0f; acc[2] = 0.0f;
}

// ---------------- kernel B: one thread per cell -----------------------------
__global__ __launch_bounds__(BLOCK)
void rl_cells(const float* __restrict__ out, const float* __restrict__ tcls,
              const float* __restrict__ cwgt,
              const float* __restrict__ tgt8, const int* __restrict__ tflat,
              float* __restrict__ acc) {
    __shared__ float s_t[NT][8];
    __shared__ int   s_flat[NT];
    __shared__ float s_red[(BLOCK / 32) * 3];

    const int b   = blockIdx.y;
    const int tid = threadIdx.x;

    if (tid < NT) {
        #pragma unroll
        for (int q = 0; q < 8; ++q) s_t[tid][q] = tgt8[(b * NT + tid) * 8 + q];
        s_flat[tid] = tflat[b * NT + tid];
    }
    __syncthreads();

    const int p   = blockIdx.x * BLOCK + tid;   // a*PLANE + j*NW + i, always < CELLS
    const int a   = p / PLANE;
    const int rem = p - a * PLANE;
    const int j   = rem / NW;
    const int i   = rem - j * NW;

    const float awv[3] = {1.28967f, 2.12714f, 3.27212f};
    const float ahv[3] = {4.15014f, 5.09344f, 5.87423f};

    const size_t base = ((size_t)b * (NAA * (5 + NC)) + (size_t)a * (5 + NC)) * PLANE + rem;
    const float o0 = out[base];
    const float o1 = out[base + 1 * PLANE];
    const float wv = out[base + 2 * PLANE];
    const float hv = out[base + 3 * PLANE];
    const float o4 = out[base + 4 * PLANE];

    const float x = sigmoidf_(o0), y = sigmoidf_(o1), conf = sigmoidf_(o4);
    const float px = x + (float)i, py = y + (float)j;
    const float pw = expf(wv) * awv[a], ph = expf(hv) * ahv[a];
    const float pxl = px - pw * 0.5f, pxr = px + pw * 0.5f;
    const float pyl = py - ph * 0.5f, pyr = py + ph * 0.5f;
    const float parea = pw * ph;

    float maxiou = 0.0f, miou = 0.0f;
    int   mt = -1;
    for (int t = 0; t < NT; ++t) {
        const int f = s_flat[t];
        if (f < 0) continue;
        const float gx = s_t[t][0], gy = s_t[t][1], gw = s_t[t][2], gh = s_t[t][3];
        const float mx = fminf(gx - gw * 0.5f, pxl);
        const float Mx = fmaxf(gx + gw * 0.5f, pxr);
        const float my = fminf(gy - gh * 0.5f, pyl);
        const float My = fmaxf(gy + gh * 0.5f, pyr);
        const float cw_ = gw + pw - (Mx - mx);
        const float ch_ = gh + ph - (My - my);
        const float inter = (cw_ <= 0.0f || ch_ <= 0.0f) ? 0.0f : cw_ * ch_;
        const float iou = inter / (gw * gh + parea - inter);
        maxiou = fmaxf(maxiou, iou);
        if (f == p) { mt = t; miou = iou; }   // last matching target wins
    }

    float conf_mask = (maxiou > 0.6f) ? 0.0f : 1.0f;   // NOOBJECT_SCALE = 1
    float coord = 0.0f, tx = 0.0f, ty = 0.0f, twt = 0.0f, tht = 0.0f, tconf = 0.0f;
    if (mt >= 0) {
        coord = 1.0f; conf_mask = 5.0f;                // OBJECT_SCALE
        tx = s_t[mt][4]; ty = s_t[mt][5]; twt = s_t[mt][6]; tht = s_t[mt][7];
        tconf = miou;
    }

    float loss = 0.5f * coord * ((x - tx) * (x - tx) + (y - ty) * (y - ty) +
                                 (wv - twt) * (wv - twt) + (hv - tht) * (hv - tht))
               + 0.5f * conf_mask * (conf - tconf) * (conf - tconf);

    // focal class loss: only matched cells contribute (coord_mask gate)
    float cls = 0.0f, nsel = 0.0f;
    if (mt >= 0) {
        nsel = 1.0f;
        const float* tc = tcls + ((size_t)b * NT + mt) * NC;
        for (int c = 0; c < NC; ++c) {
            const float pc = sigmoidf_(out[base + (size_t)(5 + c) * PLANE]);
            const float cw_ = cwgt[c];
            const float l1 = sqrtf(1.0f - pc + EPSF) * logf(pc + EPSF) * expf(cw_);
            const float l0 = sqrtf(pc + EPSF) * logf(1.0f - pc + EPSF) * expf(1.0f - cw_);
            const float tv = tc[c];
            cls += tv * l1 + (1.0f - tv) * l0;
        }
    }

    // --- full-wave reductions on the matrix pipe (EXEC all-ones here) -------
    const float wl = wave_sum_wmma(loss);
    const float wc = wave_sum_wmma(cls);
    const float wn = wave_sum_wmma(nsel);

    const int wave = tid >> 5, lane = tid & 31;
    if (lane == 0) {
        s_red[wave * 3 + 0] = wl;
        s_red[wave * 3 + 1] = wc;
        s_red[wave * 3 + 2] = wn;
    }
    __syncthreads();
    if (tid == 0) {
        float L = 0.0f, C = 0.0f, N = 0.0f;
        #pragma unroll
        for (int w2 = 0; w2 < BLOCK / 32; ++w2) {
            L += s_red[w2 * 3 + 0]; C += s_red[w2 * 3 + 1]; N += s_red[w2 * 3 + 2];
        }
        atomicAdd(&acc[0], L);
        atomicAdd(&acc[1], C);
        atomicAdd(&acc[2], N);
    }
}

// ---------------- finalize --------------------------------------------------
__global__ void rl_final(const float* __restrict__ acc, float* __restrict__ out) {
    // loss_cls = CLASS_SCALE * (-cls_sum / max(nsel, 1))
    out[0] = acc[0] - acc[1] / fmaxf(acc[2], 1.0f);
}

extern "C" void kernel_launch(void* const* d_in, const int* in_sizes, int n_in,
                              void* d_out, int out_size, void* d_ws, size_t ws_size,
                              hipStream_t stream) {
    const float* output    = (const float*)d_in[0];   // (32, 255, 56, 56)
    const float* tboxes    = (const float*)d_in[1];   // (32, 50, 4)
    const float* tcls      = (const float*)d_in[2];   // (32, 50, 80)
    const float* cwgt      = (const float*)d_in[3];   // (80,)
    float* outp = (float*)d_out;

    // workspace layout: [0..7] accumulators (3 used) | tgt8 (1600*8 f32) | tflat (1600 i32)
    float* wsf   = (float*)d_ws;
    float* acc   = wsf;
    float* tgt8  = wsf + 8;
    int*   tflat = (int*)(wsf + 8 + NB * NT * 8);

    rl_targets<<<(NB * NT + 255) / 256, 256, 0, stream>>>(tboxes, tgt8, tflat);
    rl_init<<<1, 1, 0, stream>>>(acc);
    dim3 grid(BLOCKS_PER_B, NB);
    rl_cells<<<grid, BLOCK, 0, stream>>>(output, tcls, cwgt, tgt8, tflat, acc);
    rl_final<<<1, 1, 0, stream>>>(acc, outp);
    (void)in_sizes; (void)n_in; (void)out_size; (void)ws_size;
}